// MedSAM3D_91036126806733
// MI455X (gfx1250) — compile-verified
//
#include <hip/hip_runtime.h>

typedef float v2f __attribute__((ext_vector_type(2)));
typedef float v8f __attribute__((ext_vector_type(8)));

#define F_CH   256
#define HW     4096   // 64*64
#define N_NB   25
#define WAVES  8
#define TILES_PER_WAVE 4
#define PIX_PER_BLOCK (WAVES * TILES_PER_WAVE * 16)   // 512
#define BLOCKS_PER_CH (HW / PIX_PER_BLOCK)            // 8

// out[f, p] = sum_n X[f, p, n] * W[f, n]
// WMMA mapping (V_WMMA_F32_16X16X4_F32, wave32):
//   A (16x4, MxK): A[i,k] = Wrow[kb + k] for all rows i (broadcast).
//     Lanes 0-15 hold K=0,1 in VGPR0,1; lanes 16-31 hold K=2,3.
//   B (4x16, KxN): B[k,j] = X[pix_j, kb + k], N = lane%16, K split likewise.
//   D[i,j] = partial out[pix_j], identical for every row i; row 0 lives in
//   VGPR0 of lanes 0..15, which is what we store.
__global__ __launch_bounds__(256) void medsam_compress_wmma(
    const float* __restrict__ X, const float* __restrict__ Wm,
    float* __restrict__ out)
{
  __shared__ __align__(16) float wrow[32];

  const int tid  = threadIdx.x;
  const int bx   = blockIdx.x;
  const int f    = bx >> 3;                 // bx / BLOCKS_PER_CH
  const int pblk = (bx & 7) * PIX_PER_BLOCK;
  const int lane = tid & 31;                // wave32
  const int wave = tid >> 5;

  // Stage this channel's weight row into LDS, zero-padded to K=28.
  if (tid < 32) wrow[tid] = (tid < N_NB) ? Wm[f * N_NB + tid] : 0.0f;
  __syncthreads();

  const int halfk = (lane >> 4) << 1;       // 0 for lanes 0-15, 2 for 16-31
  const int m     = lane & 15;

  // Hoist the 7 A-matrix fragments (weights, broadcast over rows) into regs.
  v2f a[7];
#pragma unroll
  for (int s = 0; s < 7; ++s) {
    a[s].x = wrow[4 * s + halfk];           // max index 26
    a[s].y = wrow[4 * s + halfk + 1];       // max index 27 (zero pad)
  }

  const int waveBase = pblk + wave * (TILES_PER_WAVE * 16);

  // One base pointer per wave; every load below uses a compile-time offset
  // from it, so addresses fold into the 24-bit instruction IOFFSET.
  const size_t rowBase = ((size_t)f * HW + waveBase + m) * N_NB;
  const float* __restrict__ rowh = X + rowBase + halfk;  // K-half folded in
  const float* __restrict__ rowm = X + rowBase;          // for the n=24 tail
  float* __restrict__ outp = out + (size_t)f * HW + waveBase + lane;

#pragma unroll
  for (int t = 0; t < TILES_PER_WAVE; ++t) {
    const int tb = t * 16 * N_NB;           // tile offset in floats (const)

    if (t + 1 < TILES_PER_WAVE)
      __builtin_prefetch(rowh + tb + 16 * N_NB, 0, 3);   // next tile, WGP scope

    v8f c = {0.f, 0.f, 0.f, 0.f, 0.f, 0.f, 0.f, 0.f};

    // 6 full K=4 steps cover n = 0..23 (strictly in-bounds loads).
#pragma unroll
    for (int s = 0; s < 6; ++s) {
      v2f b;
      b.x = rowh[tb + 4 * s];               // constant IOFFSET
      b.y = rowh[tb + 4 * s + 1];
      c = __builtin_amdgcn_wmma_f32_16x16x4_f32(
              /*neg_a=*/false, a[s], /*neg_b=*/false, b,
              /*c_mod=*/(short)0, c, /*reuse_a=*/false, /*reuse_b=*/false);
    }

    // Tail step: only K-slot 0 (n=24) is live; W[25..27]==0 kills the rest.
    // Load unconditionally (in-bounds for every lane), then select — no
    // EXEC-divergent branch around the load.
    const float r24 = rowm[tb + 24];
    v2f bt;
    bt.x = (halfk == 0) ? r24 : 0.0f;
    bt.y = 0.0f;
    c = __builtin_amdgcn_wmma_f32_16x16x4_f32(
            false, a[6], false, bt, (short)0, c, false, false);

    // D row 0: VGPR0 of lanes 0..15 holds out[waveBase + t*16 + lane].
    if (lane < 16)
      outp[t * 16] = c[0];
  }
}

extern "C" void kernel_launch(void* const* d_in, const int* in_sizes, int n_in,
                              void* d_out, int out_size, void* d_ws, size_t ws_size,
                              hipStream_t stream) {
  (void)in_sizes; (void)n_in; (void)out_size; (void)d_ws; (void)ws_size;
  const float* X  = (const float*)d_in[0];   // (1, 256, 64, 64, 25) f32
  const float* Wm = (const float*)d_in[1];   // (256, 25) f32
  float* out      = (float*)d_out;           // (1, 256, 64, 64) f32

  dim3 grid(F_CH * BLOCKS_PER_CH);           // 2048 blocks
  dim3 block(256);                           // 8 waves (wave32)
  hipLaunchKernelGGL(medsam_compress_wmma, grid, block, 0, stream, X, Wm, out);
}